// WavelengthDependentD2NNModel_38208029065723
// MI455X (gfx1250) — compile-verified
//
#include <hip/hip_runtime.h>
#include <math.h>

// ---------------------------------------------------------------------------
// Wavelength-dependent D2NN, fully fused single-workgroup kernel for gfx1250.
// B=16, C=3, H=W=5, L=5 layers.  48 images of 5x5 complex.
// FFT2(U) = F*U*F with F the symmetric 5-point DFT matrix; implemented as
// "right-multiply by 5x5 complex matrix" stages over a 240x5 row-stacked
// matrix with transposed LDS re-reads between stages.
// V_WMMA_F32_16X16X4_F32, K padded 5->8, complex = 4 real chains
// => 8 WMMAs / wave / stage, 24 stages.
// 480 threads = 15 wave32 = 15 M=16 tiles covering all 240 rows.
// B-operands (DFT columns) are per-lane constants held in registers.
// ---------------------------------------------------------------------------

typedef float v2f __attribute__((ext_vector_type(2)));
typedef float v8f __attribute__((ext_vector_type(8)));

#define NT 480

__device__ __forceinline__ v8f wmma4(v2f a, v2f b, v8f c) {
  // D = A(16x4) * B(4x16) + C   (f32)
  return __builtin_amdgcn_wmma_f32_16x16x4_f32(
      /*neg_a=*/false, a, /*neg_b=*/false, b,
      /*c_mod=*/(short)0, c, /*reuse_a=*/false, /*reuse_b=*/false);
}

// One stage: out = in * (Br + i*Bi) per 5x5 image; TR selects transposed read.
// nI* must hold -Bi.  Must be called by all 480 threads, uniform control flow.
template <bool TR>
__device__ __forceinline__ void dft_stage(
    const float* __restrict__ inRe, const float* __restrict__ inIm,
    float* __restrict__ outRe, float* __restrict__ outIm,
    v2f bRl, v2f bRh, v2f bIl, v2f bIh, v2f nIl, v2f nIh, float scale)
{
  const int t    = threadIdx.x;
  const int wave = t >> 5;        // 0..14 -> M-tile
  const int lane = t & 31;
  const int m    = lane & 15;     // A row within tile / D column
  const int half = lane >> 4;

  const int g    = wave * 16 + m;          // global row 0..239
  const int img  = g / 5;
  const int rl   = g - img * 5;
  const int base = img * 25;
  const int k0   = half ? 2 : 0;

  const int i0 = TR ? (base + (k0    ) * 5 + rl) : (base + rl * 5 + k0    );
  const int i1 = TR ? (base + (k0 + 1) * 5 + rl) : (base + rl * 5 + k0 + 1);
  const int i4 = TR ? (base + 20 + rl)           : (base + rl * 5 + 4);

  v2f aRl = { inRe[i0], inRe[i1] };
  v2f aIl = { inIm[i0], inIm[i1] };
  const float r4 = inRe[i4];               // unconditional load + select
  const float q4 = inIm[i4];
  v2f aRh = { half ? 0.f : r4, 0.f };      // K=4 (K=5..7 zero-padded)
  v2f aIh = { half ? 0.f : q4, 0.f };

  // complex matmul: Re = Ar*Br - Ai*Bi ; Im = Ar*Bi + Ai*Br
  v8f accR = { 0.f, 0.f, 0.f, 0.f, 0.f, 0.f, 0.f, 0.f };
  v8f accI = { 0.f, 0.f, 0.f, 0.f, 0.f, 0.f, 0.f, 0.f };
  accR = wmma4(aRl, bRl, accR);
  accR = wmma4(aRh, bRh, accR);
  accR = wmma4(aIl, nIl, accR);
  accR = wmma4(aIh, nIh, accR);
  accI = wmma4(aRl, bIl, accI);
  accI = wmma4(aRh, bIh, accI);
  accI = wmma4(aIl, bRl, accI);
  accI = wmma4(aIh, bRh, accI);

  // store D: VGPR j holds row (j + 8*half), column m
  if (m < 5) {
#pragma unroll
    for (int j = 0; j < 8; ++j) {
      const int grow = wave * 16 + half * 8 + j;
      const int oim  = grow / 5;
      const int orl  = grow - oim * 5;
      outRe[oim * 25 + orl * 5 + m] = accR[j] * scale;
      outIm[oim * 25 + orl * 5 + m] = accI[j] * scale;
    }
  }
}

// Per-channel sum of |z|^2: 160 threads (5 full waves) per channel,
// per-thread partials -> wave32 shuffle reduce -> one ds-atomic per wave.
__device__ __forceinline__ void energy_reduce(
    const float* __restrict__ re, const float* __restrict__ im, float* acc3)
{
  const int t = threadIdx.x;
  const int c = t / 160;          // channel, uniform within each wave
  const int j = t - c * 160;      // 0..159
  float s = 0.f;
  for (int e = j; e < 400; e += 160) {
    const int b   = e / 25;
    const int pos = e - b * 25;
    const int i   = (3 * b + c) * 25 + pos;
    s += re[i] * re[i] + im[i] * im[i];
  }
#pragma unroll
  for (int off = 16; off > 0; off >>= 1) s += __shfl_xor(s, off, 32);
  if ((t & 31) == 0) atomicAdd(&acc3[c], s);
}

__global__ __launch_bounds__(NT) void d2nn_kernel(
    const float* __restrict__ x,       // [16,3,5,5]
    const float* __restrict__ phases,  // [5,5,5]
    const float* __restrict__ coeffs,  // [5,3]
    float* __restrict__ out)           // [16,3,5,5]
{
  __shared__ float fRe[1200], fIm[1200];   // field
  __shared__ float aRe[1200], aIm[1200];   // ping
  __shared__ float bRe[1200], bIm[1200];   // pong
  __shared__ float tfR[75], tfI[75];       // transfer function per channel
  __shared__ float mult[15];               // clamped coeffs [L][C]
  __shared__ float eIn[3], eOut[3], ratio[3];

  const int t = threadIdx.x;

  // ---- per-lane constant B operands (DFT matrix columns) -----------------
  {
  }
  const int lane = t & 31;
  const int mm   = lane & 15;
  const int hh   = lane >> 4;
  const int kk0  = hh ? 2 : 0;
  float c0 = 0.f, c1 = 0.f, c4 = 0.f, s0 = 0.f, s1 = 0.f, s4 = 0.f;
  if (mm < 5) {
    const float W5 = 1.2566370614359172f;  // 2*pi/5
    const float a0 = W5 * (float)(((kk0    ) * mm) % 5);
    const float a1 = W5 * (float)(((kk0 + 1) * mm) % 5);
    c0 = cosf(a0); s0 = sinf(a0);
    c1 = cosf(a1); s1 = sinf(a1);
    if (!hh) {
      const float a4 = W5 * (float)((4 * mm) % 5);
      c4 = cosf(a4); s4 = sinf(a4);
    }
  }
  const v2f bRl = { c0,  c1 }, bRh = { c4,  0.f };
  const v2f fIl = { -s0, -s1 }, fIh = { -s4, 0.f };  // forward Bi = -sin
  const v2f gIl = { s0,  s1 }, gIh = { s4,  0.f };   // inverse Bi = +sin

  // ---- one-time LDS setup ------------------------------------------------
  if (t < 75) {
    const float freqs[5] = { 0.f, 25000.f, 50000.f, -50000.f, -25000.f }; // fftfreq(5, 8e-6)
    const float lam[3]   = { 4.5e-7f, 5.32e-7f, 6.33e-7f };
    const int c = t / 25, pos = t - c * 25, r = pos / 5, cc = pos - r * 5;
    const float f2 = freqs[r] * freqs[r] + freqs[cc] * freqs[cc];
    const float il = 1.f / lam[c];
    const float k2 = il * il - f2;
    if (k2 > 0.f) {
      const float ph = 0.05f * (6.2831853071795864f * sqrtf(k2)); // z * kz
      tfR[t] = cosf(ph);
      tfI[t] = sinf(ph);
    } else {
      tfR[t] = 0.f;
      tfI[t] = 0.f;
    }
  }
  if (t < 15) {
    float v = coeffs[t];
    v = fminf(fmaxf(v, 0.5f), 1.5f);
    if ((t % 3) == 1) v = 1.0f;   // base wavelength coefficient pinned to 1
    mult[t] = v;
  }
  for (int i = t; i < 1200; i += NT) { fRe[i] = x[i]; fIm[i] = 0.f; }
  __syncthreads();

  // ---- 5 diffraction layers ---------------------------------------------
  for (int l = 0; l < 5; ++l) {
    if (t < 3) eIn[t] = 0.f;
    __syncthreads();
    energy_reduce(fRe, fIm, eIn);   // |xc|^2 == |field|^2 (unit-modulus phase)

    // phase multiply -> (aRe, aIm)
    for (int i = t; i < 1200; i += NT) {
      const int img = i / 25;
      const int c   = img % 3;
      const int pos = i - img * 25;
      const float th = phases[l * 25 + pos] * mult[l * 3 + c];
      float ss, cs;
      sincosf(th, &ss, &cs);
      const float re = fRe[i], im = fIm[i];
      aRe[i] = re * cs - im * ss;
      aIm[i] = re * ss + im * cs;
    }
    __syncthreads();

    // FFT2: two row stages (transpose between via read pattern)
    dft_stage<false>(aRe, aIm, bRe, bIm, bRl, bRh, fIl, fIh, gIl, gIh, 1.f);
    __syncthreads();
    dft_stage<true >(bRe, bIm, aRe, aIm, bRl, bRh, fIl, fIh, gIl, gIh, 1.f);
    __syncthreads();

    // transfer-function multiply (TF symmetric -> valid on transposed data)
    for (int i = t; i < 1200; i += NT) {
      const int img = i / 25;
      const int c   = img % 3;
      const int pos = i - img * 25;
      const float tr = tfR[c * 25 + pos], ti = tfI[c * 25 + pos];
      const float re = aRe[i], im = aIm[i];
      aRe[i] = re * tr - im * ti;
      aIm[i] = re * ti + im * tr;
    }
    __syncthreads();

    // IFFT2: conjugate-DFT stages (Bi and -Bi swapped), 1/25 in last store
    dft_stage<false>(aRe, aIm, bRe, bIm, bRl, bRh, gIl, gIh, fIl, fIh, 1.f);
    __syncthreads();
    dft_stage<true >(bRe, bIm, fRe, fIm, bRl, bRh, gIl, gIh, fIl, fIh, 0.04f);
    if (t < 3) eOut[t] = 0.f;
    __syncthreads();

    energy_reduce(fRe, fIm, eOut);
    __syncthreads();
    if (t < 3) {
      const float eo = eOut[t] * (1.f / 400.f);
      const float ei = eIn[t] * (1.f / 400.f);
      float r = 1.f;
      if (eo > 1e-8f) {
        r = sqrtf(ei / eo);
        r = fminf(fmaxf(r, 0.5f), 2.f);
      }
      ratio[t] = r;
    }
    __syncthreads();
    for (int i = t; i < 1200; i += NT) {
      const float r = ratio[(i / 25) % 3];
      fRe[i] *= r;
      fIm[i] *= r;
    }
    __syncthreads();
  }

  // ---- final angular-spectrum propagation (same z = 0.05) ----------------
  dft_stage<false>(fRe, fIm, bRe, bIm, bRl, bRh, fIl, fIh, gIl, gIh, 1.f);
  __syncthreads();
  dft_stage<true >(bRe, bIm, aRe, aIm, bRl, bRh, fIl, fIh, gIl, gIh, 1.f);
  __syncthreads();
  for (int i = t; i < 1200; i += NT) {
    const int img = i / 25;
    const int c   = img % 3;
    const int pos = i - img * 25;
    const float tr = tfR[c * 25 + pos], ti = tfI[c * 25 + pos];
    const float re = aRe[i], im = aIm[i];
    aRe[i] = re * tr - im * ti;
    aIm[i] = re * ti + im * tr;
  }
  __syncthreads();
  dft_stage<false>(aRe, aIm, bRe, bIm, bRl, bRh, gIl, gIh, fIl, fIh, 1.f);
  __syncthreads();
  dft_stage<true >(bRe, bIm, fRe, fIm, bRl, bRh, gIl, gIh, fIl, fIh, 0.04f);
  __syncthreads();

  for (int i = t; i < 1200; i += NT)
    out[i] = fRe[i] * fRe[i] + fIm[i] * fIm[i];
}

extern "C" void kernel_launch(void* const* d_in, const int* in_sizes, int n_in,
                              void* d_out, int out_size, void* d_ws, size_t ws_size,
                              hipStream_t stream) {
  (void)in_sizes; (void)n_in; (void)d_ws; (void)ws_size; (void)out_size;
  const float* x      = (const float*)d_in[0];
  const float* phases = (const float*)d_in[1];
  const float* coeffs = (const float*)d_in[2];
  float* out = (float*)d_out;
  // Single workgroup: whole dependent pipeline lives in the LDS of one WGP.
  d2nn_kernel<<<dim3(1), dim3(NT), 0, stream>>>(x, phases, coeffs, out);
}